// AssemblyGNN_10445360463974
// MI455X (gfx1250) — compile-verified
//
#include <hip/hip_runtime.h>
#include <hip/hip_bf16.h>

// GCN on MI455X (gfx1250).
// - Dense X@W on v_wmma_f32_16x16x4_f32 (f32-exact, matches reference).
// - Edge aggregation converted from scatter-atomics to gather-form CSR SpMM
//   (atomic RMW throughput, not HBM bandwidth, would otherwise dominate).
// - Pooling as deterministic segment reduction over the sorted batch vector.

#define DDIM 128
#define LDA  132     // LDS row stride (pad to avoid bank conflicts)
#define NGRAPH 64
#define POOL_SPLIT 8

typedef __attribute__((ext_vector_type(2))) float v2f;
typedef __attribute__((ext_vector_type(8))) float v8f;

// ---------------------------------------------------------------------------
// GEMM: out[M x 128] = act(A)[M x 128] @ W[128 x 128] (+bias) (+relu)
// One block = 8 waves. Block stages a 128-row chunk of A in LDS.
// Each wave owns one 16-col tile; preloads its 128x16 W panel into 32 v2f
// register fragments; 32 chained v_wmma_f32_16x16x4_f32 per 16-row strip.
// ---------------------------------------------------------------------------
__global__ __launch_bounds__(256) void gcn_gemm_wmma(
    const float* __restrict__ A, const float* __restrict__ W,
    const float* __restrict__ bias, float* __restrict__ out,
    int M, int reluIn, int reluOut)
{
    __shared__ float As[128 * LDA];

    const int tid  = threadIdx.x;
    const int wave = tid >> 5;
    const int lane = tid & 31;
    const int half = lane >> 4;     // 0: lanes 0-15, 1: lanes 16-31
    const int l16  = lane & 15;
    const int rowBase = blockIdx.x * 128;

    // Cooperative stage of 128x128 activation chunk (ReLU fused on input).
    for (int i = tid; i < 128 * 32; i += 256) {
        int r  = i >> 5;            // row in chunk (32 float4 per row)
        int c4 = (i & 31) << 2;     // starting col
        float4 v = make_float4(0.f, 0.f, 0.f, 0.f);
        if (rowBase + r < M)
            v = *(const float4*)(A + (size_t)(rowBase + r) * DDIM + c4);
        if (reluIn) {
            v.x = fmaxf(v.x, 0.f); v.y = fmaxf(v.y, 0.f);
            v.z = fmaxf(v.z, 0.f); v.w = fmaxf(v.w, 0.f);
        }
        float* d = As + r * LDA + c4;
        d[0] = v.x; d[1] = v.y; d[2] = v.z; d[3] = v.w;
    }
    __syncthreads();

    const int n0 = wave * 16;   // this wave's output column tile

    // Preload B panel: for k-block kb, B(4x16) frag:
    //   VGPR0 = row (2*half + 0), VGPR1 = row (2*half + 1)  [ISA 7.12.2]
    v2f bfrag[32];
#pragma unroll
    for (int kb = 0; kb < 32; ++kb) {
        int k0 = kb * 4 + half * 2;
        v2f b;
        b.x = W[(size_t)(k0    ) * DDIM + n0 + l16];
        b.y = W[(size_t)(k0 + 1) * DDIM + n0 + l16];
        bfrag[kb] = b;
    }

    for (int s = 0; s < 8; ++s) {
        int m0 = s * 16;
        if (rowBase + m0 >= M) break;       // uniform per wave: EXEC stays full
        v8f c = {};
#pragma unroll
        for (int kb = 0; kb < 32; ++kb) {
            int k0 = kb * 4 + half * 2;
            v2f a;
            // A(16x4): lanes 0-15 hold K=0,1; lanes 16-31 hold K=2,3
            a.x = As[(m0 + l16) * LDA + k0];
            a.y = As[(m0 + l16) * LDA + k0 + 1];
            c = __builtin_amdgcn_wmma_f32_16x16x4_f32(
                    false, a, false, bfrag[kb], (short)0, c, false, false);
        }
        // C/D 16x16: VGPR v holds rows (v + 8*half), col = n0 + l16
#pragma unroll
        for (int v = 0; v < 8; ++v) {
            int m = rowBase + m0 + v + half * 8;
            if (m < M) {
                float val = c[v];
                if (bias)    val += bias[n0 + l16];
                if (reluOut) val = fmaxf(val, 0.f);
                out[(size_t)m * DDIM + n0 + l16] = val;
            }
        }
    }
}

// ---------------------------------------------------------------------------
// CSR build: count in-edges per dst, exclusive scan, scatter edge payloads.
// Degree with self-loop falls out of the row lengths: deg = count + 1.
// ---------------------------------------------------------------------------
__global__ void gcn_csr_zero(int* __restrict__ cursor, int N) {
    int i = blockIdx.x * blockDim.x + threadIdx.x;
    if (i < N) cursor[i] = 0;
}
__global__ void gcn_csr_count(const int* __restrict__ ei, int* __restrict__ cursor, int E) {
    int e = blockIdx.x * blockDim.x + threadIdx.x;
    if (e < E) atomicAdd(&cursor[ei[E + e]], 1);   // dst row of edge_index
}

// Single-block exclusive scan over N counts (N <= 1024*64 here N=50000).
// Also emits dinv[i] = rsqrt(1 + count) and resets cursor to row starts.
__global__ __launch_bounds__(1024) void gcn_csr_scan(
    int* __restrict__ start, int* __restrict__ cursor,
    float* __restrict__ dinv, int N, int E)
{
    __shared__ int partial[1024];
    const int t = threadIdx.x;
    const int C = (N + 1023) / 1024;
    const int lo = t * C;
    const int hi = min(lo + C, N);

    int sum = 0;
    for (int i = lo; i < hi; ++i) sum += cursor[i];
    partial[t] = sum;
    __syncthreads();

    for (int off = 1; off < 1024; off <<= 1) {     // inclusive Hillis-Steele
        int v = partial[t];
        int add = (t >= off) ? partial[t - off] : 0;
        __syncthreads();
        partial[t] = v + add;
        __syncthreads();
    }

    int base = (t == 0) ? 0 : partial[t - 1];
    for (int i = lo; i < hi; ++i) {
        int c = cursor[i];
        start[i]  = base;
        cursor[i] = base;
        dinv[i]   = rsqrtf(1.0f + (float)c);       // deg includes self-loop
        base += c;
    }
    if (t == 0) start[N] = E;
}

__global__ void gcn_csr_fill(const int* __restrict__ ei, const float* __restrict__ dinv,
                             int* __restrict__ cursor, int* __restrict__ csr_src,
                             float* __restrict__ csr_w, int E)
{
    int e = blockIdx.x * blockDim.x + threadIdx.x;
    if (e < E) {
        int s  = ei[e];
        int dn = ei[E + e];
        int slot = atomicAdd(&cursor[dn], 1);
        csr_src[slot] = s;
        csr_w[slot]   = dinv[s] * dinv[dn];
    }
}

// ---------------------------------------------------------------------------
// Gather-form aggregation: each dst row sums its in-edges in registers,
// one coalesced store per output dword (no float atomics).
// agg[n] = bias + dinv[n]^2 * hw[n] + sum_j w_j * hw[src_j]
// ---------------------------------------------------------------------------
__global__ void gcn_agg_csr(const float* __restrict__ hw, const float* __restrict__ dinv,
                            const int* __restrict__ start, const int* __restrict__ csr_src,
                            const float* __restrict__ csr_w, const float* __restrict__ bias,
                            float* __restrict__ agg, int N)
{
    int idx = blockIdx.x * blockDim.x + threadIdx.x;  // 128 threads per node
    int n = idx >> 7, d = idx & 127;
    if (n < N) {
        float di  = dinv[n];
        float acc = bias[d] + di * di * hw[idx];      // self-loop (norm = 1/deg)
        int s0 = start[n], s1 = start[n + 1];         // uniform across the group
        for (int j = s0; j < s1; ++j) {
            int   s = csr_src[j];
            float w = csr_w[j];
            acc += w * hw[(size_t)s * DDIM + d];      // 512B-coalesced gather
        }
        agg[idx] = acc;
    }
}

// ---------------------------------------------------------------------------
// Deterministic global mean pool over the SORTED batch vector.
// ---------------------------------------------------------------------------
__device__ __forceinline__ int gcn_lower_bound(const int* __restrict__ b, int N, int val) {
    int lo = 0, hi = N;
    while (lo < hi) { int m = (lo + hi) >> 1; if (b[m] < val) lo = m + 1; else hi = m; }
    return lo;
}

__global__ __launch_bounds__(128) void gcn_pool_partial(
    const float* __restrict__ h, const int* __restrict__ batch,
    float* __restrict__ part, int N)
{
    int g = blockIdx.x / POOL_SPLIT;
    int k = blockIdx.x % POOL_SPLIT;
    int d = threadIdx.x;
    int lo  = gcn_lower_bound(batch, N, g);
    int hi  = gcn_lower_bound(batch, N, g + 1);
    int len = hi - lo;
    int c0 = lo + (int)(((long long)len * k)       / POOL_SPLIT);
    int c1 = lo + (int)(((long long)len * (k + 1)) / POOL_SPLIT);
    float acc = 0.f;
    for (int n = c0; n < c1; ++n) acc += h[(size_t)n * DDIM + d];
    part[(size_t)(g * POOL_SPLIT + k) * DDIM + d] = acc;
}

__global__ __launch_bounds__(128) void gcn_pool_final(
    const float* __restrict__ part, const int* __restrict__ batch,
    float* __restrict__ pooled, int N)
{
    int g = blockIdx.x;
    int d = threadIdx.x;
    int lo = gcn_lower_bound(batch, N, g);
    int hi = gcn_lower_bound(batch, N, g + 1);
    float acc = 0.f;
#pragma unroll
    for (int k = 0; k < POOL_SPLIT; ++k)           // fixed order: deterministic
        acc += part[(size_t)(g * POOL_SPLIT + k) * DDIM + d];
    pooled[(size_t)g * DDIM + d] = acc / fmaxf((float)(hi - lo), 1.0f);
}

// ---------------------------------------------------------------------------
extern "C" void kernel_launch(void* const* d_in, const int* in_sizes, int n_in,
                              void* d_out, int out_size, void* d_ws, size_t ws_size,
                              hipStream_t stream) {
    const float* x   = (const float*)d_in[0];
    const float* W1  = (const float*)d_in[1];
    const float* b1  = (const float*)d_in[2];
    const float* W2  = (const float*)d_in[3];
    const float* b2  = (const float*)d_in[4];
    const float* W3  = (const float*)d_in[5];
    const float* b3  = (const float*)d_in[6];
    const float* Wm1 = (const float*)d_in[7];
    const float* bm1 = (const float*)d_in[8];
    const float* Wm2 = (const float*)d_in[9];
    const float* bm2 = (const float*)d_in[10];
    const int*   ei    = (const int*)d_in[11];
    const int*   batch = (const int*)d_in[12];

    const int N = in_sizes[0] / DDIM;     // 50000
    const int E = in_sizes[11] / 2;       // 800000

    // workspace layout (all regions 16B-aligned; ~59 MB total)
    char* p = (char*)d_ws;
    int*   start   = (int*)p;   p += (size_t)(N + 4) * 4;        // N+1 used, pad
    int*   cursor  = (int*)p;   p += (size_t)N * 4;
    int*   csr_src = (int*)p;   p += (size_t)E * 4;
    float* csr_w   = (float*)p; p += (size_t)E * 4;
    float* dinv    = (float*)p; p += (size_t)N * 4;
    float* hw      = (float*)p; p += (size_t)N * DDIM * 4;       // GEMM out / MLP t1
    float* agg     = (float*)p; p += (size_t)N * DDIM * 4;       // aggregated h
    float* part    = (float*)p; p += (size_t)NGRAPH * POOL_SPLIT * DDIM * 4;
    float* pooled  = (float*)p;

    const int nThreads = 256;
    const int gN    = (N + nThreads - 1) / nThreads;
    const int gE    = (E + nThreads - 1) / nThreads;
    const int gND   = (N * DDIM + nThreads - 1) / nThreads;
    const int gGemm = (N + 127) / 128;

    // CSR of incoming edges + deg^{-1/2} (built once, reused by all 3 layers)
    gcn_csr_zero <<<gN, nThreads, 0, stream>>>(cursor, N);
    gcn_csr_count<<<gE, nThreads, 0, stream>>>(ei, cursor, E);
    gcn_csr_scan <<<1, 1024, 0, stream>>>(start, cursor, dinv, N, E);
    gcn_csr_fill <<<gE, nThreads, 0, stream>>>(ei, dinv, cursor, csr_src, csr_w, E);

    // layer 1
    gcn_gemm_wmma<<<gGemm, nThreads, 0, stream>>>(x,   W1, nullptr, hw, N, 0, 0);
    gcn_agg_csr  <<<gND,  nThreads, 0, stream>>>(hw, dinv, start, csr_src, csr_w, b1, agg, N);
    // layer 2 (ReLU of previous output fused into GEMM input stage)
    gcn_gemm_wmma<<<gGemm, nThreads, 0, stream>>>(agg, W2, nullptr, hw, N, 1, 0);
    gcn_agg_csr  <<<gND,  nThreads, 0, stream>>>(hw, dinv, start, csr_src, csr_w, b2, agg, N);
    // layer 3 (no output ReLU)
    gcn_gemm_wmma<<<gGemm, nThreads, 0, stream>>>(agg, W3, nullptr, hw, N, 1, 0);
    gcn_agg_csr  <<<gND,  nThreads, 0, stream>>>(hw, dinv, start, csr_src, csr_w, b3, agg, N);

    // deterministic global mean pool (batch is sorted)
    gcn_pool_partial<<<NGRAPH * POOL_SPLIT, 128, 0, stream>>>(agg, batch, part, N);
    gcn_pool_final  <<<NGRAPH, 128, 0, stream>>>(part, batch, pooled, N);

    // MLP head: out = relu(pooled @ Wm1 + bm1) @ Wm2 + bm2   (M = 64, same WMMA kernel)
    gcn_gemm_wmma<<<1, nThreads, 0, stream>>>(pooled, Wm1, bm1, hw, NGRAPH, 0, 1);
    gcn_gemm_wmma<<<1, nThreads, 0, stream>>>(hw, Wm2, bm2, (float*)d_out, NGRAPH, 0, 0);
}